// FlaxAttentionModule_52673478918586
// MI455X (gfx1250) — compile-verified
//
#include <hip/hip_runtime.h>
#include <float.h>

// Native clang vector: legal for __builtin_nontemporal_* and lowers to b128 vmem ops.
typedef float v4f __attribute__((ext_vector_type(4)));

// Exact vector type the async-LDS builtins expect (from hipcc's diagnostic:
// parameter is "__attribute__((__vector_size__(4*sizeof(int)))) int __device__*").
typedef int v4i __attribute__((vector_size(16)));
typedef __attribute__((address_space(1))) v4i* gptr_t;   // global v4i*
typedef __attribute__((address_space(3))) v4i* lptr_t;   // LDS v4i*

// Problem constants (match the reference harness exactly).
#define B_ 4
#define Q_ 128
#define H_ 32
#define D_ 128
#define L_ 4096

// Everything below in v4f (16B) units — all powers of two.
#define F4_PER_ROW   ((H_ * D_) / 4)          // 1024  = 2^10  (one (h,d) row)
#define F4_PER_BATCH ((L_ * H_ * D_) / 4)     // 4194304 = 2^22
#define F4_TOTAL     (B_ * F4_PER_BATCH)      // 16777216
#define F4_NEW_BATCH ((Q_ * H_ * D_) / 4)     // 131072

#if __has_builtin(__builtin_amdgcn_global_load_async_to_lds_b128) && \
    __has_builtin(__builtin_amdgcn_global_store_async_from_lds_b128) && \
    __has_builtin(__builtin_amdgcn_s_wait_asynccnt)
#define HAS_ASYNC_LDS 1
#else
#define HAS_ASYNC_LDS 0
#endif

// Per-lane source selection for the fused dynamic_update_slice:
// rows l in [ci, ci+Q) come from the new K/V tile, all others from the cache.
__device__ __forceinline__ const v4f* select_src(const v4f* __restrict__ cache,
                                                 const v4f* __restrict__ newkv,
                                                 long long idx, int ci) {
    const int b     = (int)(idx >> 22);
    const int l     = (int)((idx >> 10) & (L_ - 1));
    const int inrow = (int)(idx & (F4_PER_ROW - 1));
    if (l >= ci && l < ci + Q_)
        return newkv + ((long long)b * F4_NEW_BATCH
                        + (long long)(l - ci) * F4_PER_ROW + inrow);
    return cache + idx;
}

// -----------------------------------------------------------------------------
// Kernel 1: fused cache copy + dynamic_update_slice (pure streaming, ~1 GB).
// Preferred path: CDNA5 async data mover — global -> LDS -> global via
// GLOBAL_LOAD_ASYNC_TO_LDS_B128 / GLOBAL_STORE_ASYNC_FROM_LDS_B128, fenced with
// s_wait_asynccnt. 4 b128 transfers in flight per lane per phase; each thread
// owns a private LDS slot so no barriers are needed (the per-wave asynccnt
// fence also protects LDS reuse, since async loads/stores are unordered
// against each other per the ISA).
// -----------------------------------------------------------------------------
__global__ void kv_update_copy(const v4f* __restrict__ cache,
                               const v4f* __restrict__ newkv,
                               v4f* __restrict__ out,
                               const int* __restrict__ cache_index_p) {
    // Uniform scalar: cache_index % L, clamped like lax.dynamic_update_slice.
    int ci = *cache_index_p;
    ci = ci % L_;
    if (ci > L_ - Q_) ci = L_ - Q_;
    const int tid = threadIdx.x;

#if HAS_ASYNC_LDS
    __shared__ v4f stage[4][256];                      // 16 KB / block

    // 1024 v4f per block-chunk; F4_TOTAL % 1024 == 0 -> every chunk is full,
    // EXEC stays all-ones, no guards needed.
    const long long chunkStride = (long long)gridDim.x * 1024;
    for (long long c0 = (long long)blockIdx.x * 1024;
         c0 < (long long)F4_TOTAL; c0 += chunkStride) {
        // Warm GL2 one chunk ahead on the dominant stream (global_prefetch_b8).
        if (c0 + chunkStride < (long long)F4_TOTAL)
            __builtin_prefetch((const void*)(cache + c0 + chunkStride + tid), 0, 0);

        #pragma unroll
        for (int j = 0; j < 4; ++j) {
            const long long idx = c0 + j * 256 + tid;
            const v4f* src = select_src(cache, newkv, idx, ci);
            __builtin_amdgcn_global_load_async_to_lds_b128(
                (gptr_t)src, (lptr_t)&stage[j][tid], 0, 0);
        }
        __builtin_amdgcn_s_wait_asynccnt(0);           // loads landed in LDS
        #pragma unroll
        for (int j = 0; j < 4; ++j) {
            const long long idx = c0 + j * 256 + tid;
            __builtin_amdgcn_global_store_async_from_lds_b128(
                (gptr_t)(out + idx), (lptr_t)&stage[j][tid], 0, 0);
        }
        __builtin_amdgcn_s_wait_asynccnt(0);           // LDS safe to overwrite
    }
#else
    // Fallback: NT b128 VGPR copy.
    long long idx = (long long)blockIdx.x * blockDim.x + tid;
    const long long stride = (long long)gridDim.x * blockDim.x;
    for (; idx < (long long)F4_TOTAL; idx += stride) {
        if (idx + stride < (long long)F4_TOTAL)
            __builtin_prefetch((const void*)(cache + idx + stride), 0, 0);
        const v4f* src = select_src(cache, newkv, idx, ci);
        v4f v = __builtin_nontemporal_load(src);
        __builtin_nontemporal_store(v, &out[idx]);
    }
#endif
}

// -----------------------------------------------------------------------------
// Kernel 2: attention bias.
// bias[b, 0, qi, l] = (attn[b,l] && causal[ci+qi, l] && l < ci+Q) ? 0 : -FLT_MAX
// One v4f (4 consecutive l) per thread; bool masks read 4-at-a-time as u32.
// -----------------------------------------------------------------------------
__global__ void bias_kernel(const unsigned char* __restrict__ attn,    // (B, L) bool
                            const unsigned char* __restrict__ causal,  // (L, L) bool
                            v4f* __restrict__ bias,                    // (B, Q, L/4)
                            const int* __restrict__ cache_index_p) {
    const int ci = *cache_index_p;
    int ci_c = ci;                       // dynamic_slice start clamp
    if (ci_c > L_ - Q_) ci_c = L_ - Q_;

    const int N = B_ * Q_ * (L_ / 4);    // 524288 v4f
    const int idx = blockIdx.x * blockDim.x + threadIdx.x;
    if (idx >= N) return;

    const int b  = idx >> 17;            // Q*L/4 = 2^17 per batch
    const int qi = (idx >> 10) & (Q_ - 1);
    const int l0 = (idx & (L_ / 4 - 1)) * 4;

    const unsigned int am4 =
        *(const unsigned int*)(attn + (long long)b * L_ + l0);
    const unsigned int cm4 =
        *(const unsigned int*)(causal + (long long)(ci_c + qi) * L_ + l0);
    const int limit = ci + Q_;           // pad_mask uses raw cache_index + q

    v4f r;
    #pragma unroll
    for (int j = 0; j < 4; ++j) {
        const bool a   = ((am4 >> (8 * j)) & 0xFFu) != 0u;
        const bool c   = ((cm4 >> (8 * j)) & 0xFFu) != 0u;
        const bool pad = (l0 + j) < limit;
        r[j] = (a && c && pad) ? 0.0f : -FLT_MAX;
    }
    __builtin_nontemporal_store(r, &bias[idx]);
}

// -----------------------------------------------------------------------------
// Launch. Inputs (setup_inputs order):
//  0 query (unused)  1 key  2 value  3 cache_key  4 cache_value
//  5 attention_mask (bool)  6 causal_mask (bool)  7 cache_index (int scalar)
// Output: key_cache | value_cache | bias, concatenated flat.
// -----------------------------------------------------------------------------
extern "C" void kernel_launch(void* const* d_in, const int* in_sizes, int n_in,
                              void* d_out, int out_size, void* d_ws, size_t ws_size,
                              hipStream_t stream) {
    const float* key = (const float*)d_in[1];
    const float* val = (const float*)d_in[2];
    const float* ck  = (const float*)d_in[3];
    const float* cv  = (const float*)d_in[4];
    const unsigned char* am = (const unsigned char*)d_in[5];
    const unsigned char* cm = (const unsigned char*)d_in[6];
    const int* ci = (const int*)d_in[7];

    float* out_k = (float*)d_out;
    float* out_v = out_k + (size_t)B_ * L_ * H_ * D_;
    float* out_b = out_v + (size_t)B_ * L_ * H_ * D_;

    const dim3 blk(256);  // 8 wave32 waves per block

    // 4096 blocks: async path -> 4 chunks of 1024 v4f per block;
    // fallback path -> 16 v4f per thread grid-stride.
    kv_update_copy<<<4096, blk, 0, stream>>>(
        (const v4f*)ck, (const v4f*)key, (v4f*)out_k, ci);
    kv_update_copy<<<4096, blk, 0, stream>>>(
        (const v4f*)cv, (const v4f*)val, (v4f*)out_v, ci);

    const int nb = (B_ * Q_ * (L_ / 4) + 255) / 256;
    bias_kernel<<<nb, blk, 0, stream>>>(am, cm, (v4f*)out_b, ci);
}